// NERModel_49495203119124
// MI455X (gfx1250) — compile-verified
//
#include <hip/hip_runtime.h>
#include <math.h>

typedef float v2f __attribute__((ext_vector_type(2)));
typedef float v8f __attribute__((ext_vector_type(8)));

#define NLAB     17
#define HDIM     384
#define NTOK     65536          // 64 * 1024
#define SEQ      1024
#define NBATCH   64
#define BIGIDX   NTOK
#define IGNORE_L (-100)
#define WPAD     388            // 384 + 4: kills LDS bank conflicts (388 % 64 == 4)

struct Ws {
    float ce_sum;  int ce_cnt;
    float ctx_sum; int ctx_cnt;
    int   d_i;
    int   m1[NLAB];             // first index with label t
    int   m2[NLAB];             // second index with label t
    int   cnt[NLAB];            // count of label t
};

__global__ void k_init(Ws* ws) {
    int t = threadIdx.x;
    if (t == 0) { ws->ce_sum = 0.f; ws->ce_cnt = 0; ws->ctx_sum = 0.f; ws->ctx_cnt = 0; ws->d_i = BIGIDX; }
    if (t < NLAB) { ws->m1[t] = BIGIDX; ws->m2[t] = BIGIDX; ws->cnt[t] = 0; }
}

__global__ __launch_bounds__(256) void k_scan1(const int* __restrict__ labels,
                                               const int* __restrict__ mask, Ws* ws) {
    int i = blockIdx.x * 256 + threadIdx.x;
    if (i >= NTOK) return;
    int l = labels[i];
    bool m = mask[i] > 0;
    if (m && l == 0) atomicMin(&ws->d_i, i);
    if (m && l > 0 && l < NLAB) { atomicMin(&ws->m1[l], i); atomicAdd(&ws->cnt[l], 1); }
}

__global__ __launch_bounds__(256) void k_scan2(const int* __restrict__ labels,
                                               const int* __restrict__ mask, Ws* ws) {
    int i = blockIdx.x * 256 + threadIdx.x;
    if (i >= NTOK) return;
    int l = labels[i];
    bool m = mask[i] > 0;
    if (m && l > 0 && l < NLAB && i != ws->m1[l]) atomicMin(&ws->m2[l], i);
}

// Main fused kernel: logits via V_WMMA_F32_16X16X4_F32, then log-softmax + NLL.
// 256 threads = 8 waves; each wave owns 16 rows -> 128 rows per block.
__global__ __launch_bounds__(256) void k_ce(const float* __restrict__ emb,
                                            const float* __restrict__ Wc,
                                            const float* __restrict__ bc,
                                            const int*   __restrict__ labels,
                                            Ws* __restrict__ ws) {
    __shared__ float sW[32 * WPAD];     // W zero-padded to 32 classes
    __shared__ float sB[32];
    __shared__ float sLog[128 * 20];    // per-row logits (17 used, pad 20)
    __shared__ float sCe;
    __shared__ int   sCnt;

    const int tid = threadIdx.x;

    for (int i = tid; i < 32 * WPAD; i += 256) sW[i] = 0.f;
    __syncthreads();
    for (int i = tid; i < NLAB * HDIM; i += 256) {
        int c = i / HDIM, h = i - c * HDIM;
        sW[c * WPAD + h] = Wc[i];
    }
    if (tid < 32) sB[tid] = (tid < NLAB) ? bc[tid] : 0.f;
    if (tid == 0) { sCe = 0.f; sCnt = 0; }
    __syncthreads();

    const int wave = tid >> 5;
    const int lane = tid & 31;
    const int hi   = lane >> 4;      // K half: 0 -> K=0,1 ; 1 -> K=2,3
    const int lc   = lane & 15;

    const int  rowBase = blockIdx.x * 128;
    const long rowA    = rowBase + wave * 16 + lc;      // A-matrix M index
    const float* aPtr  = emb + rowA * HDIM + 2 * hi;
    const float* b0Ptr = &sW[lc * WPAD + 2 * hi];       // B cols 0..15
    const float* b1Ptr = &sW[(16 + lc) * WPAD + 2 * hi];// B cols 16..31 (>=17 zero)

    v8f acc0 = {};
    v8f acc1 = {};

#pragma unroll 4
    for (int k = 0; k < HDIM; k += 4) {
        v2f a  = *(const v2f*)(aPtr + k);
        v2f b0 = *(const v2f*)(b0Ptr + k);
        v2f b1 = *(const v2f*)(b1Ptr + k);
        acc0 = __builtin_amdgcn_wmma_f32_16x16x4_f32(false, a, false, b0,
                                                     (short)0, acc0, false, false);
        acc1 = __builtin_amdgcn_wmma_f32_16x16x4_f32(false, a, false, b1,
                                                     (short)0, acc1, false, false);
    }

    // D layout: vgpr v, lanes 0-15 -> row v, col lane; lanes 16-31 -> row v+8.
#pragma unroll
    for (int v = 0; v < 8; ++v) {
        int rl = wave * 16 + hi * 8 + v;
        sLog[rl * 20 + lc] = acc0[v];
        if (lc == 0) sLog[rl * 20 + 16] = acc1[v];   // class 16 is col 0 of tile 1
    }
    __syncthreads();

    if (tid < 128) {
        int n   = rowBase + tid;
        int lbl = labels[n];
        float x[NLAB];
        float mx = -3.0e38f;
#pragma unroll
        for (int c = 0; c < NLAB; ++c) {
            x[c] = sLog[tid * 20 + c] + sB[c];
            mx = fmaxf(mx, x[c]);
        }
        float se = 0.f;
#pragma unroll
        for (int c = 0; c < NLAB; ++c) se += __expf(x[c] - mx);
        if (lbl != IGNORE_L) {
            int sl = (lbl >= 0 && lbl < NLAB) ? lbl : 0;
            float nll = (mx + __logf(se)) - x[sl];
            atomicAdd(&sCe, nll);
            atomicAdd(&sCnt, 1);
        }
    }
    __syncthreads();
    if (tid == 0) {
        atomicAdd(&ws->ce_sum, sCe);
        atomicAdd(&ws->ce_cnt, sCnt);
    }
}

// ctx loss: one wave per adjacent-token pair, early-out on label predicate.
__global__ __launch_bounds__(256) void k_ctx(const float* __restrict__ emb,
                                             const int* __restrict__ labels, Ws* ws) {
    int wave = threadIdx.x >> 5, lane = threadIdx.x & 31;
    long p = (long)blockIdx.x * 8 + wave;
    if (p >= (long)NBATCH * (SEQ - 1)) return;
    int  b = (int)(p / (SEQ - 1));
    int  s = (int)(p % (SEQ - 1));
    long n = (long)b * SEQ + s;
    int l0 = labels[n], l1 = labels[n + 1];
    if (!(l0 != IGNORE_L && l0 == l1 && l0 > 0)) return;
    const float* e0 = emb + n * HDIM;
    const float* e1 = e0 + HDIM;
    float acc = 0.f;
    for (int h = lane; h < HDIM; h += 32) {
        float d = e0[h] - e1[h];
        acc += d * d;
    }
#pragma unroll
    for (int off = 16; off > 0; off >>= 1) acc += __shfl_xor(acc, off, 32);
    if (lane == 0) {
        atomicAdd(&ws->ctx_sum, acc * (1.0f / HDIM));
        atomicAdd(&ws->ctx_cnt, 1);
    }
}

// quad loss + final combine: 16 waves, one per entity type.
__global__ __launch_bounds__(512) void k_final(const float* __restrict__ emb,
                                               Ws* ws, float* __restrict__ out) {
    __shared__ float sQl[16];
    __shared__ int   sOk[16];
    int tid = threadIdx.x, wave = tid >> 5, lane = tid & 31;
    int t = wave + 1;                       // type 1..16

    int ai = ws->m1[t], pi = ws->m2[t], ct = ws->cnt[t], di = ws->d_i;
    int ni = BIGIDX;
    for (int tt = 1; tt < NLAB; ++tt)
        if (tt != t) ni = min(ni, ws->m1[tt]);
    bool ok = (ct >= 2) && (ni < BIGIDX) && (di < BIGIDX);

    long ca = min(ai, NTOK - 1), cp = min(pi, NTOK - 1);
    long cn = min(ni, NTOK - 1), cd = min(di, NTOK - 1);

    float spd = 0.f, snd = 0.f, sdd = 0.f;
    for (int h = lane; h < HDIM; h += 32) {
        float A  = emb[ca * HDIM + h];
        float P  = emb[cp * HDIM + h];
        float Ng = emb[cn * HDIM + h];
        float D  = emb[cd * HDIM + h];
        float d1 = A - P  + 1e-6f; spd += d1 * d1;
        float d2 = A - Ng + 1e-6f; snd += d2 * d2;
        float d3 = A - D  + 1e-6f; sdd += d3 * d3;
    }
#pragma unroll
    for (int off = 16; off > 0; off >>= 1) {
        spd += __shfl_xor(spd, off, 32);
        snd += __shfl_xor(snd, off, 32);
        sdd += __shfl_xor(sdd, off, 32);
    }
    if (lane == 0) {
        float pd = sqrtf(spd), nd = sqrtf(snd), dd = sqrtf(sdd);
        float ql = fmaxf(pd - nd + 1.0f, 0.f) + fmaxf(pd - dd + 2.0f, 0.f);
        sQl[wave] = ok ? ql : 0.f;
        sOk[wave] = ok ? 1 : 0;
    }
    __syncthreads();
    if (tid == 0) {
        float qs = 0.f; int qc = 0;
        for (int w = 0; w < 16; ++w) { qs += sQl[w]; qc += sOk[w]; }
        float quad = (qc > 0) ? qs / (float)qc : 0.f;
        float ce   = ws->ce_sum / (float)max(ws->ce_cnt, 1);
        float ctx  = (ws->ctx_cnt > 0) ? ws->ctx_sum / (float)ws->ctx_cnt : 0.f;
        out[0] = ce + 0.5f * quad + 0.1f * ctx;
    }
}

extern "C" void kernel_launch(void* const* d_in, const int* in_sizes, int n_in,
                              void* d_out, int out_size, void* d_ws, size_t ws_size,
                              hipStream_t stream) {
    (void)in_sizes; (void)n_in; (void)out_size; (void)ws_size;
    const float* emb    = (const float*)d_in[0];
    const float* Wc     = (const float*)d_in[1];
    const float* bc     = (const float*)d_in[2];
    const int*   labels = (const int*)d_in[3];
    const int*   mask   = (const int*)d_in[4];
    float* out = (float*)d_out;
    Ws*    ws  = (Ws*)d_ws;

    k_init <<<1, 64, 0, stream>>>(ws);
    k_scan1<<<(NTOK + 255) / 256, 256, 0, stream>>>(labels, mask, ws);
    k_scan2<<<(NTOK + 255) / 256, 256, 0, stream>>>(labels, mask, ws);
    k_ce   <<<NTOK / 128, 256, 0, stream>>>(emb, Wc, bc, labels, ws);
    {
        long pairs = (long)NBATCH * (SEQ - 1);
        int blocks = (int)((pairs + 7) / 8);
        k_ctx<<<blocks, 256, 0, stream>>>(emb, labels, ws);
    }
    k_final<<<1, 512, 0, stream>>>(emb, ws, out);
}